// Sampler_1726576855245
// MI455X (gfx1250) — compile-verified
//
#include <hip/hip_runtime.h>
#include <hip/hip_bf16.h>

// ---------------------------------------------------------------------------
// Gemma-2 sampler head for MI455X (gfx1250, wave32).
//   d_in[0] embedding      [V=128000, D=2048] f32
//   d_in[1] hidden_states  [B=32, S=8, D=2048] f32
//   d_in[2] output_positions [1] i32
//   d_in[3] temperatures   [B] f32
//   d_in[4] top_ps         [B] f32
//   d_in[5] top_ks         [B] i32
//   d_out = [token_ids (32 f32)] ++ [logits (32*128000 f32)]
// ---------------------------------------------------------------------------

#define V_TOKENS  128000
#define DMODEL    2048
#define BATCH     32
#define SEQ       8
#define SOFTCAP_F 30.0f

#define KC        256          // K-chunk staged in LDS per iteration
#define LDS_STRIDE 260         // padded dword row stride (260%64==4 -> conflict-free)
#define WAVES     8
#define TPB       (WAVES * 32)
#define NTILE     (WAVES * 16) // vocab rows per workgroup

typedef __attribute__((ext_vector_type(2))) float v2f;
typedef __attribute__((ext_vector_type(8))) float v8f;
typedef __attribute__((ext_vector_type(4))) int   v4i;

typedef __attribute__((address_space(1))) v4i* gptr_v4i;
typedef __attribute__((address_space(3))) v4i* lptr_v4i;

#if defined(__has_builtin)
#  if __has_builtin(__builtin_amdgcn_global_load_async_to_lds_b128)
#    define HAS_ASYNC_LDS 1
#  endif
#  if __has_builtin(__builtin_amdgcn_s_wait_asynccnt)
#    define HAS_WAIT_ASYNC 1
#  endif
#endif

// ---------------------------------------------------------------------------
// Kernel 1: logits = tanh((hs @ E^T)/30)*30 / temp   via V_WMMA_F32_16X16X4_F32
// ---------------------------------------------------------------------------
__global__ __launch_bounds__(TPB) void gemm_logits_kernel(
    const float* __restrict__ emb,       // [V, D]
    const float* __restrict__ hidden,    // [B, S, D]
    const int*   __restrict__ out_pos,   // [1]
    const float* __restrict__ temps,     // [B]
    float*       __restrict__ logits_out // [B, V]
) {
  __shared__ float lds_hs[BATCH * LDS_STRIDE];

  const int tid   = threadIdx.x;
  const int wave  = tid >> 5;
  const int lane  = tid & 31;
  const int ln    = lane & 15;           // N index within 16-wide tile / M within 16
  const int khalf = (lane >> 4) << 1;    // lanes 0-15 -> K{0,1}; lanes 16-31 -> K{2,3}
  const int pos   = out_pos[0];
  const int v0    = blockIdx.x * NTILE + wave * 16;

  v8f acc0 = {};  // batch rows 0..15
  v8f acc1 = {};  // batch rows 16..31

  // staging decomposition: 32 rows x 256 floats per chunk; 8 threads/row, 8 float4 each
  const int srow = tid >> 3;   // 0..31
  const int sseg = tid & 7;    // 0..7

  for (int kc = 0; kc < DMODEL; kc += KC) {
    const float* src = hidden + ((size_t)srow * SEQ + pos) * DMODEL + kc + sseg * 32;
    float*       dst = lds_hs + srow * LDS_STRIDE + sseg * 32;
#if defined(HAS_ASYNC_LDS)
#pragma unroll
    for (int j = 0; j < 8; ++j) {
      __builtin_amdgcn_global_load_async_to_lds_b128(
          (gptr_v4i)(src + j * 4),
          (lptr_v4i)(dst + j * 4),
          0, 0);
    }
#  if defined(HAS_WAIT_ASYNC)
    __builtin_amdgcn_s_wait_asynccnt(0);
#  else
    asm volatile("s_wait_asynccnt 0" ::: "memory");
#  endif
#else
#pragma unroll
    for (int j = 0; j < 8; ++j) {
      float4 x = ((const float4*)src)[j];
      ((float4*)dst)[j] = x;
    }
#endif
    __syncthreads();

    // B fragment stream: lane ln covers vocab row v0+ln; khalf selects K pair
    const float* ebase  = emb + (size_t)(v0 + ln) * DMODEL + kc + khalf;
    const float* a0base = lds_hs + ln * LDS_STRIDE + khalf;
    const float* a1base = lds_hs + (16 + ln) * LDS_STRIDE + khalf;

#pragma unroll 4
    for (int k4 = 0; k4 < KC / 4; ++k4) {
      v2f bfrag = *(const v2f*)(ebase + k4 * 4);
      v2f a0    = *(const v2f*)(a0base + k4 * 4);
      v2f a1    = *(const v2f*)(a1base + k4 * 4);
      acc0 = __builtin_amdgcn_wmma_f32_16x16x4_f32(
          false, a0, false, bfrag, (short)0, acc0, false, false);
      acc1 = __builtin_amdgcn_wmma_f32_16x16x4_f32(
          false, a1, false, bfrag, (short)0, acc1, false, false);
    }
    __syncthreads();
  }

  // Epilogue: C/D layout -> VGPR r, lanes 0-15 hold M=r, lanes 16-31 hold M=r+8
  const float invcap = 1.0f / SOFTCAP_F;
  const int mbase = (lane >> 4) << 3;  // 0 or 8
#pragma unroll
  for (int r = 0; r < 8; ++r) {
    const int b0 = mbase + r;
    const int b1 = b0 + 16;
    float x0 = tanhf(acc0[r] * invcap) * SOFTCAP_F / temps[b0];
    float x1 = tanhf(acc1[r] * invcap) * SOFTCAP_F / temps[b1];
    logits_out[(size_t)b0 * V_TOKENS + v0 + ln] = x0;
    logits_out[(size_t)b1 * V_TOKENS + v0 + ln] = x1;
  }
}

// ---------------------------------------------------------------------------
// Kernel 2: softmax + top-p/top-k + categorical draw.  top_k < 64, so only the
// top-64 probabilities per row can ever survive the masks.
// ---------------------------------------------------------------------------
#define NEG_BIG (-1e30f)

__global__ __launch_bounds__(256) void sample_kernel(
    const float* __restrict__ logits,   // [B, V]
    const float* __restrict__ top_ps,   // [B]
    const int*   __restrict__ top_ks,   // [B]
    float*       __restrict__ token_out // [B]
) {
  __shared__ float s_val[1024];
  __shared__ int   s_idx[1024];
  __shared__ float r_v[256];
  __shared__ int   r_i[256];
  __shared__ float s_top_v[64];
  __shared__ int   s_top_i[64];
  __shared__ float s_gmax, s_Z;

  const int b   = blockIdx.x;
  const int tid = threadIdx.x;
  const float* row = logits + (size_t)b * V_TOKENS;

  // pass 1: per-thread top-4 over strided slice
  float c0 = NEG_BIG, c1 = NEG_BIG, c2 = NEG_BIG, c3 = NEG_BIG;
  int   j0 = 0, j1 = 0, j2 = 0, j3 = 0;
  for (int v = tid; v < V_TOKENS; v += 256) {
    float x = row[v];
    if (x > c3) {
      if (x > c0)      { c3=c2;j3=j2; c2=c1;j2=j1; c1=c0;j1=j0; c0=x;j0=v; }
      else if (x > c1) { c3=c2;j3=j2; c2=c1;j2=j1; c1=x;j1=v; }
      else if (x > c2) { c3=c2;j3=j2; c2=x;j2=v; }
      else             { c3=x;j3=v; }
    }
  }
  s_val[tid*4+0] = c0; s_idx[tid*4+0] = j0;
  s_val[tid*4+1] = c1; s_idx[tid*4+1] = j1;
  s_val[tid*4+2] = c2; s_idx[tid*4+2] = j2;
  s_val[tid*4+3] = c3; s_idx[tid*4+3] = j3;

  // row max reduction
  r_v[tid] = c0;
  __syncthreads();
  for (int s = 128; s > 0; s >>= 1) {
    if (tid < s && r_v[tid + s] > r_v[tid]) r_v[tid] = r_v[tid + s];
    __syncthreads();
  }
  if (tid == 0) s_gmax = r_v[0];
  __syncthreads();
  const float gmax = s_gmax;

  // pass 2: sum of exp(x - gmax)
  float lsum = 0.0f;
  for (int v = tid; v < V_TOKENS; v += 256) lsum += __expf(row[v] - gmax);
  r_v[tid] = lsum;
  __syncthreads();
  for (int s = 128; s > 0; s >>= 1) {
    if (tid < s) r_v[tid] += r_v[tid + s];
    __syncthreads();
  }
  if (tid == 0) s_Z = r_v[0];
  __syncthreads();
  const float Z = s_Z;

  // extract top-64 candidates in descending order
  for (int it = 0; it < 64; ++it) {
    float bv = NEG_BIG; int bi = 0;
#pragma unroll
    for (int j = 0; j < 4; ++j) {
      int c = tid + j * 256;
      if (s_val[c] > bv) { bv = s_val[c]; bi = c; }
    }
    r_v[tid] = bv; r_i[tid] = bi;
    __syncthreads();
    for (int s = 128; s > 0; s >>= 1) {
      if (tid < s && r_v[tid + s] > r_v[tid]) { r_v[tid] = r_v[tid + s]; r_i[tid] = r_i[tid + s]; }
      __syncthreads();
    }
    if (tid == 0) {
      int w = r_i[0];
      s_top_v[it] = r_v[0];
      s_top_i[it] = s_idx[w];
      s_val[w] = NEG_BIG;
    }
    __syncthreads();
  }

  // thread 0: top-p (exclusive cumsum) + top-k masks, renormalize, inverse-CDF draw
  if (tid == 0) {
    const float tp = top_ps[b];
    const int   tk = top_ks[b];
    float p[64];
    float cum = 0.0f, tot = 0.0f;
    for (int j = 0; j < 64; ++j) {
      float pj = __expf(s_top_v[j] - gmax) / Z;
      float cum_ex = cum;
      cum += pj;
      if (cum_ex > tp) pj = 0.0f;
      if (j >= tk)     pj = 0.0f;
      p[j] = pj;
      tot += pj;
    }
    // hash-based uniform (PCG-style), keyed on sample key 42 and row
    unsigned st = 42u ^ ((unsigned)b * 0x9E3779B9u);
    st = st * 747796405u + 2891336453u;
    unsigned wrd = ((st >> ((st >> 28) + 4u)) ^ st) * 277803737u;
    wrd = (wrd >> 22) ^ wrd;
    float u = (float)(wrd >> 8) * (1.0f / 16777216.0f) * tot;
    float acc = 0.0f;
    int sel = 0;
    for (int j = 0; j < 64; ++j) {
      acc += p[j];
      if (p[j] > 0.0f) { sel = j; if (u <= acc) break; }
    }
    token_out[b] = (float)s_top_i[sel];
  }
}

// ---------------------------------------------------------------------------
extern "C" void kernel_launch(void* const* d_in, const int* in_sizes, int n_in,
                              void* d_out, int out_size, void* d_ws, size_t ws_size,
                              hipStream_t stream) {
  const float* emb     = (const float*)d_in[0];
  const float* hidden  = (const float*)d_in[1];
  const int*   out_pos = (const int*)  d_in[2];
  const float* temps   = (const float*)d_in[3];
  const float* top_ps  = (const float*)d_in[4];
  const int*   top_ks  = (const int*)  d_in[5];

  float* out    = (float*)d_out;
  float* logits = out + BATCH;   // tokens first, then logits, in return order

  gemm_logits_kernel<<<V_TOKENS / NTILE, TPB, 0, stream>>>(
      emb, hidden, out_pos, temps, logits);

  sample_kernel<<<BATCH, 256, 0, stream>>>(logits, top_ps, top_ks, out);
}